// UOTPooling_60619168416422
// MI455X (gfx1250) — compile-verified
//
#include <hip/hip_runtime.h>
#include <math.h>

#define B_SEG 1024
#define NMAX  128
#define DDIM  256
#define HDIM  512
#define NUM_IT 4
#define EPSQ  1e-8f
#define NEGV  -1e9f
#define LOGD  5.545177444479562f   // ln(256)

typedef float v2f __attribute__((ext_vector_type(2)));
typedef float v8f __attribute__((ext_vector_type(8)));

__device__ __forceinline__ float softplusf(float v) {
    return (v > 20.f) ? v : log1pf(expf(v));
}

__device__ __forceinline__ float wave_max(float v) {
    #pragma unroll
    for (int off = 16; off > 0; off >>= 1) v = fmaxf(v, __shfl_xor(v, off, 32));
    return v;
}
__device__ __forceinline__ float wave_sum(float v) {
    #pragma unroll
    for (int off = 16; off > 0; off >>= 1) v += __shfl_xor(v, off, 32);
    return v;
}

// ---------------------------------------------------------------------------
// Kernel 1: logits[m] = sum_c tanh( (x @ W1^T)[m,c] ) * W2[c]
// fp32 WMMA 16x16x4. One block = one 16-row M tile; 8 waves x 4 N-tiles = 512 cols.
// Out-of-range tile rows are clamped to row 0 (their results are never stored),
// so the K loop has NO predication: EXEC is all-1s at every WMMA and the A/B
// fragments are plain b64 loads.
// ---------------------------------------------------------------------------
__global__ __launch_bounds__(256) void k_gemm_logits(
    const float* __restrict__ x, const float* __restrict__ W1,
    const float* __restrict__ W2, float* __restrict__ logits, int T)
{
    __shared__ float acc[16];
    const int t = threadIdx.x, lane = t & 31, wv = t >> 5;
    if (t < 16) acc[t] = 0.f;
    __syncthreads();

    const int m0 = blockIdx.x * 16;
    const int mr = m0 + (lane & 15);
    const int mclamp = (mr < T) ? mr : 0;           // always-valid row
    const int kh = (lane >> 4) * 2;                 // K sub-offset: 0 or 2
    const float* xr = x + (size_t)mclamp * DDIM + kh;
    const float* wr = W1 + (size_t)(wv * 16 + (lane & 15)) * DDIM + kh;

    v8f c[4];
    #pragma unroll
    for (int q = 0; q < 4; ++q) c[q] = v8f{0.f,0.f,0.f,0.f,0.f,0.f,0.f,0.f};

    #pragma unroll 2
    for (int k0 = 0; k0 < DDIM; k0 += 4) {
        const v2f a = *(const v2f*)(xr + k0);       // A 16x4 f32 frag (2 VGPRs)
        #pragma unroll
        for (int q = 0; q < 4; ++q) {
            const v2f bb = *(const v2f*)(wr + (size_t)q * 128 * DDIM + k0);
            c[q] = __builtin_amdgcn_wmma_f32_16x16x4_f32(
                       false, a, false, bb, (short)0, c[q], false, false);
        }
    }

    // fold tanh + W2 dot into per-row LDS accumulators
    #pragma unroll
    for (int q = 0; q < 4; ++q) {
        const int col = wv * 16 + q * 128 + (lane & 15);
        const float w2 = W2[col];
        #pragma unroll
        for (int r8 = 0; r8 < 8; ++r8) {
            const float h = tanhf(c[q][r8]);
            atomicAdd(&acc[r8 + 8 * (lane >> 4)], h * w2);
        }
    }
    __syncthreads();
    if (t < 16 && (m0 + t) < T) logits[m0 + t] = acc[t];
}

// ---------------------------------------------------------------------------
// Segment bookkeeping
// ---------------------------------------------------------------------------
__global__ void k_zero_counts(int* counts) {
    int i = blockIdx.x * blockDim.x + threadIdx.x;
    if (i < B_SEG) counts[i] = 0;
}
__global__ void k_count(const int* __restrict__ batch, int* counts, int T) {
    int i = blockIdx.x * blockDim.x + threadIdx.x;
    if (i < T) atomicAdd(&counts[batch[i]], 1);
}
__global__ __launch_bounds__(1024) void k_scan(const int* __restrict__ counts,
                                               int* __restrict__ starts) {
    __shared__ int s[B_SEG];
    int t = threadIdx.x;
    s[t] = counts[t];
    __syncthreads();
    for (int off = 1; off < B_SEG; off <<= 1) {
        int v = (t >= off) ? s[t - off] : 0;
        __syncthreads();
        s[t] += v;
        __syncthreads();
    }
    starts[t] = s[t] - counts[t];   // exclusive prefix
}

// ---------------------------------------------------------------------------
// Kernel 2: per-segment softmax -> q0
// ---------------------------------------------------------------------------
__global__ __launch_bounds__(128) void k_q0(
    const float* __restrict__ logits, const int* __restrict__ starts,
    const int* __restrict__ counts, float* __restrict__ q0)
{
    const int b = blockIdx.x, t = threadIdx.x;
    const int st = starts[b], n = counts[b];
    __shared__ float red[128];
    const float l = (t < n) ? logits[st + t] : -3.0e38f;
    red[t] = l; __syncthreads();
    for (int off = 64; off; off >>= 1) {
        if (t < off) red[t] = fmaxf(red[t], red[t + off]);
        __syncthreads();
    }
    const float m = red[0];
    __syncthreads();
    const float e = (t < n) ? expf(l - m) : 0.f;
    red[t] = e; __syncthreads();
    for (int off = 64; off; off >>= 1) {
        if (t < off) red[t] += red[t + off];
        __syncthreads();
    }
    const float s = red[0];
    if (t < n) q0[st + t] = e / (s + 1e-16f);
}

// ---------------------------------------------------------------------------
// Kernel 3: fused badmm (4 iterations) + output reduction.
// One block per batch element. 1024 threads = 32 waves.
// Wave w owns rows 4w..4w+3; lane l owns cols {l+32j}. State (log_t, log_s, z)
// lives in registers (4x8 each). Row reductions = wave shuffles; column
// reductions via 32KB LDS partials. x is re-read from L2 (fits in 192MB).
// ---------------------------------------------------------------------------
__global__ __launch_bounds__(1024, 1) void k_badmm(
    const float* __restrict__ x, const float* __restrict__ q0,
    const int* __restrict__ starts, const int* __restrict__ counts,
    const float* __restrict__ rho, const float* __restrict__ a1,
    const float* __restrict__ a2, const float* __restrict__ a3,
    float* __restrict__ out)
{
    __shared__ float red[32 * 256];     // per-wave column partials (32 KB)
    __shared__ float colmax[256];
    __shared__ float collse[256];
    __shared__ float log_mu[256];
    __shared__ float z1v[256];
    __shared__ float lq0v[128];
    __shared__ float log_eta[128];
    __shared__ float z2v[128];

    const int b = blockIdx.x;
    const int t = threadIdx.x, lane = t & 31, wv = t >> 5;
    const int st = starts[b], n = counts[b];

    if (t < 128) {
        const float q = (t < n) ? q0[st + t] : 0.f;
        const float lq = logf(q + EPSQ);
        lq0v[t] = lq; log_eta[t] = lq; z2v[t] = 0.f;
    }
    if (t < 256) { log_mu[t] = -LOGD; z1v[t] = 0.f; }
    __syncthreads();

    float lt[4][8], ls[4][8], zz[4][8];
    #pragma unroll
    for (int rr = 0; rr < 4; ++rr) {
        const float lq = lq0v[4 * wv + rr];
        #pragma unroll
        for (int j = 0; j < 8; ++j) {
            lt[rr][j] = lq - LOGD; ls[rr][j] = lq - LOGD; zz[rr][j] = 0.f;
        }
    }

    for (int k = 0; k < NUM_IT; ++k) {
        const float r  = softplusf(rho[k]);
        const float b1 = softplusf(a1[k]);
        const float b2 = softplusf(a2[k]);
        const float b3 = softplusf(a3[k]);
        const float inv2 = 1.f / (b2 + r);
        const float inv1 = 1.f / (b1 + r);
        const float inv3 = 1.f / (b3 + r);

        // ---- T-update: row (over D) log-softmax; y kept in lt regs
        #pragma unroll
        for (int rr = 0; rr < 4; ++rr) {
            const int i = 4 * wv + rr;
            const bool valid = (i < n);
            const float* xr = x + (size_t)(st + (valid ? i : 0)) * DDIM;
            float m = -3.0e38f;
            #pragma unroll
            for (int j = 0; j < 8; ++j) {
                const int d = lane + 32 * j;
                const float xv = valid ? xr[d] : 0.f;
                const float y = (xv - zz[rr][j] + r * ls[rr][j]) * inv2;
                lt[rr][j] = y;
                m = fmaxf(m, y);
            }
            m = wave_max(m);
            float se = 0.f;
            #pragma unroll
            for (int j = 0; j < 8; ++j) se += expf(lt[rr][j] - m);
            se = wave_sum(se);
            const float lse = m + logf(se);
            const float le = log_eta[i];
            #pragma unroll
            for (int j = 0; j < 8; ++j) lt[rr][j] = le + lt[rr][j] - lse;
        }

        // ---- S-update: column (over N) log-softmax; y2 kept in ls regs
        #pragma unroll
        for (int rr = 0; rr < 4; ++rr) {
            const int i = 4 * wv + rr;
            const float ng = (i < n) ? 0.f : NEGV;
            #pragma unroll
            for (int j = 0; j < 8; ++j)
                ls[rr][j] = (zz[rr][j] + r * lt[rr][j]) * inv1 + ng;
        }
        #pragma unroll
        for (int j = 0; j < 8; ++j) {
            float m = -3.0e38f;
            #pragma unroll
            for (int rr = 0; rr < 4; ++rr) m = fmaxf(m, ls[rr][j]);
            red[wv * 256 + lane + 32 * j] = m;
        }
        __syncthreads();
        if (t < 256) {
            float m = -3.0e38f;
            for (int w = 0; w < 32; ++w) m = fmaxf(m, red[w * 256 + t]);
            colmax[t] = m;
        }
        __syncthreads();
        #pragma unroll
        for (int j = 0; j < 8; ++j) {
            const int d = lane + 32 * j;
            const float cm = colmax[d];
            float s = 0.f;
            #pragma unroll
            for (int rr = 0; rr < 4; ++rr) s += expf(ls[rr][j] - cm);
            red[wv * 256 + d] = s;
        }
        __syncthreads();
        if (t < 256) {
            float s = 0.f;
            for (int w = 0; w < 32; ++w) s += red[w * 256 + t];
            collse[t] = colmax[t] + logf(s);
        }
        __syncthreads();
        #pragma unroll
        for (int j = 0; j < 8; ++j) {
            const int d = lane + 32 * j;
            const float lm = log_mu[d], cl = collse[d];
            #pragma unroll
            for (int rr = 0; rr < 4; ++rr) ls[rr][j] = lm + ls[rr][j] - cl;
        }
        __syncthreads();   // all reads of old log_mu done

        // ---- marginal updates (lse_N(log_s)==log_mu, lse_D(log_t)==log_eta)
        if (t < 256) log_mu[t] = (b3 * (-LOGD) - z1v[t] + r * log_mu[t]) * inv3;
        if (lane == 0) {
            #pragma unroll
            for (int rr = 0; rr < 4; ++rr) {
                const int i = 4 * wv + rr;
                log_eta[i] = (b3 * lq0v[i] - z2v[i] + r * log_eta[i]) * inv3;
            }
        }

        // ---- dual ascent
        float sp[8];
        #pragma unroll
        for (int j = 0; j < 8; ++j) sp[j] = 0.f;
        #pragma unroll
        for (int rr = 0; rr < 4; ++rr) {
            const int i = 4 * wv + rr;
            const float mk = (i < n) ? 1.f : 0.f;
            float rsum = 0.f;
            #pragma unroll
            for (int j = 0; j < 8; ++j) {
                const float tv = expf(lt[rr][j]) * mk;
                const float sv = expf(ls[rr][j]) * mk;
                zz[rr][j] += r * (tv - sv);
                rsum += tv;
                sp[j] += sv;
            }
            rsum = wave_sum(rsum);
            if (lane == 0)
                z2v[i] += r * (expf(log_eta[i]) - rsum) * mk;
        }
        #pragma unroll
        for (int j = 0; j < 8; ++j) red[wv * 256 + lane + 32 * j] = sp[j];
        __syncthreads();
        if (t < 256) {
            float s = 0.f;
            for (int w = 0; w < 32; ++w) s += red[w * 256 + t];
            z1v[t] += r * (expf(log_mu[t]) - s);
        }
        __syncthreads();   // end of iteration: protect red/log_eta/log_mu
    }

    // ---- output: out[b,d] = sum_i D * x[i,d] * exp(log_t[i,d]) * mask[i]
    float op[8];
    #pragma unroll
    for (int j = 0; j < 8; ++j) op[j] = 0.f;
    #pragma unroll
    for (int rr = 0; rr < 4; ++rr) {
        const int i = 4 * wv + rr;
        const bool valid = (i < n);
        const float* xr = x + (size_t)(st + (valid ? i : 0)) * DDIM;
        const float mk = valid ? (float)DDIM : 0.f;
        #pragma unroll
        for (int j = 0; j < 8; ++j) {
            const int d = lane + 32 * j;
            const float xv = valid ? xr[d] : 0.f;
            op[j] += mk * xv * expf(lt[rr][j]);
        }
    }
    #pragma unroll
    for (int j = 0; j < 8; ++j) red[wv * 256 + lane + 32 * j] = op[j];
    __syncthreads();
    if (t < 256) {
        float s = 0.f;
        for (int w = 0; w < 32; ++w) s += red[w * 256 + t];
        out[(size_t)b * DDIM + t] = s;
    }
}

// ---------------------------------------------------------------------------
extern "C" void kernel_launch(void* const* d_in, const int* in_sizes, int n_in,
                              void* d_out, int out_size, void* d_ws, size_t ws_size,
                              hipStream_t stream) {
    const float* x     = (const float*)d_in[0];
    const int*   batch = (const int*)  d_in[1];
    const float* W1    = (const float*)d_in[2];
    const float* W2    = (const float*)d_in[3];
    const float* rho   = (const float*)d_in[4];
    const float* a1    = (const float*)d_in[5];
    const float* a2    = (const float*)d_in[6];
    const float* a3    = (const float*)d_in[7];
    const int T = in_sizes[0] / DDIM;
    float* out = (float*)d_out;

    float* logits = (float*)d_ws;         // T floats
    float* q0     = logits + T;           // T floats
    int*   counts = (int*)(q0 + T);       // B ints
    int*   starts = counts + B_SEG;       // B ints

    k_zero_counts<<<(B_SEG + 255) / 256, 256, 0, stream>>>(counts);
    k_count<<<(T + 255) / 256, 256, 0, stream>>>(batch, counts, T);
    k_scan<<<1, 1024, 0, stream>>>(counts, starts);
    k_gemm_logits<<<(T + 15) / 16, 256, 0, stream>>>(x, W1, W2, logits, T);
    k_q0<<<B_SEG, 128, 0, stream>>>(logits, starts, counts, q0);
    k_badmm<<<B_SEG, 1024, 0, stream>>>(x, q0, starts, counts,
                                        rho, a1, a2, a3, out);
}